// BILSTMCRF_12919261626813
// MI455X (gfx1250) — compile-verified
//
#include <hip/hip_runtime.h>
#include <math.h>

// ---------------- configuration ----------------
#define BLOCK 256
#define KSTEPS 8
#define CHUNK (BLOCK * KSTEPS)   // 2048 steps per block
#define NEGF (-1.0e30f)

// ---------------- CDNA5 async global->LDS path ----------------
#if defined(__has_builtin)
#if __has_builtin(__builtin_amdgcn_global_load_async_to_lds_b128)
#define HAVE_ASYNC_LDS 1
#endif
#endif

#ifdef HAVE_ASYNC_LDS
typedef int v4i_cp __attribute__((ext_vector_type(4)));
typedef __attribute__((address_space(1))) v4i_cp* gptr_b128;
typedef __attribute__((address_space(3))) v4i_cp* lptr_b128;

__device__ __forceinline__ void async_cp16(const void* g, void* l) {
    __builtin_amdgcn_global_load_async_to_lds_b128(
        (gptr_b128)g, (lptr_b128)l, /*offset=*/0, /*cpol=*/0);
}
__device__ __forceinline__ void wait_async0() {
#if __has_builtin(__builtin_amdgcn_s_wait_asynccnt)
    __builtin_amdgcn_s_wait_asynccnt(0);
#else
    asm volatile("s_wait_asynccnt 0" ::: "memory");
#endif
}
#endif

// ---------------- log-semiring helpers ----------------
// Hardware transcendentals: v_exp_f32 / v_log_f32. d <= 0 always, so
// __expf never overflows and underflows cleanly to 0 (identity-safe).
__device__ __forceinline__ float lse2(float a, float b) {
    float m = fmaxf(a, b);
    float d = fminf(a, b) - m;
    return m + __logf(1.0f + __expf(d));
}

// combine (ordered): C = A o B, C[i][j] = LSE_k(A[i][k] + B[k][j])
#define COMBINE(R00,R01,R10,R11,o00,o01,o10,o11)                    \
    do {                                                            \
        float n00 = lse2((R00) + (o00), (R01) + (o10));             \
        float n01 = lse2((R00) + (o01), (R01) + (o11));             \
        float n10 = lse2((R10) + (o00), (R11) + (o10));             \
        float n11 = lse2((R10) + (o01), (R11) + (o11));             \
        (R00) = n00; (R01) = n01; (R10) = n10; (R11) = n11;         \
    } while (0)

// ---------------- kernel 1: per-block partials ----------------
__global__ __launch_bounds__(BLOCK)
void crf_partial_kernel(const float* __restrict__ em,      // [T,2]
                        const int*   __restrict__ label,   // [T]
                        const int*   __restrict__ wstate,  // [T] in 0..2
                        const int*   __restrict__ pstate,  // [T] in 0..19
                        const float* __restrict__ wpar,    // [2,2,2] = 8 floats
                        const float* __restrict__ ppar,    // [19,2,2] = 76 floats
                        float*       __restrict__ part)    // [NB,6]
{
    __shared__ float4 s_em4[CHUNK / 2];     // CHUNK*2 floats = 16 KB
    __shared__ int4   s_lab4[CHUNK / 4];    // 8 KB
    __shared__ int4   s_w4[CHUNK / 4];      // 8 KB
    __shared__ int4   s_p4[CHUNK / 4];      // 8 KB
    __shared__ float4 s_w2w[3];             // padded who2who table (row 2 = 0)
    __shared__ float4 s_pos[20];            // padded position table (row 19 = 0)
    __shared__ float  s_red[8][8];          // per-wave partials

    const int tid = threadIdx.x;
    const long long chunk = (long long)blockIdx.x * CHUNK;

    const float4* gem  = (const float4*)(em + chunk * 2);
    const int4*   glab = (const int4*)(label + chunk);
    const int4*   gw   = (const int4*)(wstate + chunk);
    const int4*   gp   = (const int4*)(pstate + chunk);

    // Compile-time trip counts -> no exec-mask branch chains around the
    // async b128 issues (every thread does all iterations).
#ifdef HAVE_ASYNC_LDS
    #pragma unroll
    for (int k = 0; k < (CHUNK / 2) / BLOCK; ++k) {       // 4 iters
        const int i = k * BLOCK + tid;
        async_cp16(&gem[i], &s_em4[i]);
    }
    #pragma unroll
    for (int k = 0; k < (CHUNK / 4) / BLOCK; ++k) {       // 2 iters
        const int i = k * BLOCK + tid;
        async_cp16(&glab[i], &s_lab4[i]);
        async_cp16(&gw[i],   &s_w4[i]);
        async_cp16(&gp[i],   &s_p4[i]);
    }
#else
    #pragma unroll
    for (int k = 0; k < (CHUNK / 2) / BLOCK; ++k) {
        const int i = k * BLOCK + tid;
        s_em4[i] = gem[i];
    }
    #pragma unroll
    for (int k = 0; k < (CHUNK / 4) / BLOCK; ++k) {
        const int i = k * BLOCK + tid;
        s_lab4[i] = glab[i];
        s_w4[i]   = gw[i];
        s_p4[i]   = gp[i];
    }
#endif

    // padded transition tables into LDS (rows w==2, p==19 are zero)
    float* pw = (float*)s_w2w;
    float* pp = (float*)s_pos;
    if (tid < 12) pw[tid] = (tid < 8)  ? wpar[tid] : 0.0f;
    if (tid < 80) pp[tid] = (tid < 76) ? ppar[tid] : 0.0f;

#ifdef HAVE_ASYNC_LDS
    wait_async0();
#endif
    __syncthreads();

    const float2* s_e2  = (const float2*)s_em4;
    const int*    s_lab = (const int*)s_lab4;
    const int*    s_w   = (const int*)s_w4;
    const int*    s_p   = (const int*)s_p4;

    const int li0 = tid * KSTEPS;

    // log-semiring identity
    float R00 = 0.0f, R01 = NEGF, R10 = NEGF, R11 = 0.0f;
    float gEm = 0.0f, gTr = 0.0f;

    int prev;
    if (li0 > 0)          prev = s_lab[li0 - 1];
    else if (chunk > 0)   prev = label[chunk - 1];
    else                  prev = 0;   // t==0: transition not used

    #pragma unroll
    for (int k = 0; k < KSTEPS; ++k) {
        const int li = li0 + k;
        const float2 e  = s_e2[li];
        const int w = s_w[li];
        const int p = s_p[li];
        const float4 a4 = s_w2w[w];
        const float4 b4 = s_pos[p];
        const float t00 = a4.x + b4.x, t01 = a4.y + b4.y;
        const float t10 = a4.z + b4.z, t11 = a4.w + b4.w;

        // R' = R o M,  M[i][j] = e_j + t_ij  (factor e_j out of the LSE)
        float c0 = lse2(R00 + t00, R01 + t10);
        float c1 = lse2(R00 + t01, R01 + t11);
        float c2 = lse2(R10 + t00, R11 + t10);
        float c3 = lse2(R10 + t01, R11 + t11);
        R00 = e.x + c0; R01 = e.y + c1;
        R10 = e.x + c2; R11 = e.y + c3;

        // gold score contributions
        const int l = s_lab[li];
        gEm += l ? e.y : e.x;
        if (chunk + li > 0) {
            gTr += (prev == 0) ? (l == 0 ? t00 : t01)
                               : (l == 0 ? t10 : t11);
        }
        prev = l;
    }

    // ordered wave32 reduction (lane order == time order)
    #pragma unroll
    for (int off = 16; off >= 1; off >>= 1) {
        float o00 = __shfl_down(R00, off), o01 = __shfl_down(R01, off);
        float o10 = __shfl_down(R10, off), o11 = __shfl_down(R11, off);
        float oe  = __shfl_down(gEm, off), ot  = __shfl_down(gTr, off);
        COMBINE(R00, R01, R10, R11, o00, o01, o10, o11);
        gEm += oe; gTr += ot;
    }

    const int wave = tid >> 5, lane = tid & 31;
    if (lane == 0) {
        s_red[wave][0] = R00; s_red[wave][1] = R01;
        s_red[wave][2] = R10; s_red[wave][3] = R11;
        s_red[wave][4] = gEm; s_red[wave][5] = gTr;
    }
    __syncthreads();

    if (tid < 8) {
        R00 = s_red[tid][0]; R01 = s_red[tid][1];
        R10 = s_red[tid][2]; R11 = s_red[tid][3];
        gEm = s_red[tid][4]; gTr = s_red[tid][5];
        #pragma unroll
        for (int off = 4; off >= 1; off >>= 1) {
            float o00 = __shfl_down(R00, off), o01 = __shfl_down(R01, off);
            float o10 = __shfl_down(R10, off), o11 = __shfl_down(R11, off);
            float oe  = __shfl_down(gEm, off), ot  = __shfl_down(gTr, off);
            COMBINE(R00, R01, R10, R11, o00, o01, o10, o11);
            gEm += oe; gTr += ot;
        }
        if (tid == 0) {
            float* o = part + (size_t)blockIdx.x * 6;
            o[0] = R00; o[1] = R01; o[2] = R10; o[3] = R11;
            o[4] = gEm; o[5] = gTr;
        }
    }
}

// ---------------- kernel 2: ordered combine of block partials ----------------
__global__ __launch_bounds__(BLOCK)
void crf_final_kernel(const float* __restrict__ part, float* __restrict__ out,
                      int nrec)
{
    __shared__ float s_red[8][8];
    const int tid = threadIdx.x;
    const int per = (nrec + BLOCK - 1) / BLOCK;
    const int base = tid * per;

    float R00 = 0.0f, R01 = NEGF, R10 = NEGF, R11 = 0.0f;
    float gEm = 0.0f, gTr = 0.0f;

    for (int k = 0; k < per; ++k) {
        const int idx = base + k;
        if (idx >= nrec) break;
        const float* q = part + (size_t)idx * 6;
        COMBINE(R00, R01, R10, R11, q[0], q[1], q[2], q[3]);
        gEm += q[4]; gTr += q[5];
    }

    #pragma unroll
    for (int off = 16; off >= 1; off >>= 1) {
        float o00 = __shfl_down(R00, off), o01 = __shfl_down(R01, off);
        float o10 = __shfl_down(R10, off), o11 = __shfl_down(R11, off);
        float oe  = __shfl_down(gEm, off), ot  = __shfl_down(gTr, off);
        COMBINE(R00, R01, R10, R11, o00, o01, o10, o11);
        gEm += oe; gTr += ot;
    }

    const int wave = tid >> 5, lane = tid & 31;
    if (lane == 0) {
        s_red[wave][0] = R00; s_red[wave][1] = R01;
        s_red[wave][2] = R10; s_red[wave][3] = R11;
        s_red[wave][4] = gEm; s_red[wave][5] = gTr;
    }
    __syncthreads();

    if (tid < 8) {
        R00 = s_red[tid][0]; R01 = s_red[tid][1];
        R10 = s_red[tid][2]; R11 = s_red[tid][3];
        gEm = s_red[tid][4]; gTr = s_red[tid][5];
        #pragma unroll
        for (int off = 4; off >= 1; off >>= 1) {
            float o00 = __shfl_down(R00, off), o01 = __shfl_down(R01, off);
            float o10 = __shfl_down(R10, off), o11 = __shfl_down(R11, off);
            float oe  = __shfl_down(gEm, off), ot  = __shfl_down(gTr, off);
            COMBINE(R00, R01, R10, R11, o00, o01, o10, o11);
            gEm += oe; gTr += ot;
        }
        if (tid == 0) {
            // alpha0 == 0  =>  total = LSE over all 4 entries of the composed map
            out[0] = gEm + gTr;                                   // gold score
            out[1] = lse2(lse2(R00, R01), lse2(R10, R11));        // partition
        }
    }
}

// ---------------- launch ----------------
extern "C" void kernel_launch(void* const* d_in, const int* in_sizes, int n_in,
                              void* d_out, int out_size, void* d_ws, size_t ws_size,
                              hipStream_t stream) {
    const float* em   = (const float*)d_in[0];  // [T,2] f32
    const int*   lab  = (const int*)d_in[1];    // [T] i32
    const int*   wst  = (const int*)d_in[2];    // [T] i32
    const int*   pst  = (const int*)d_in[3];    // [T] i32
    const float* wpar = (const float*)d_in[4];  // [2,2,2] f32
    const float* ppar = (const float*)d_in[5];  // [19,2,2] f32

    const int T  = in_sizes[1];
    const int NB = T / CHUNK;                   // T = 2^21 -> 1024 blocks

    float* part = (float*)d_ws;                 // NB * 6 floats = 24 KB

    crf_partial_kernel<<<NB, BLOCK, 0, stream>>>(em, lab, wst, pst, wpar, ppar, part);
    crf_final_kernel<<<1, BLOCK, 0, stream>>>(part, (float*)d_out, NB);
}